// kernel_generated_5_62904091017387
// MI455X (gfx1250) — compile-verified
//
#include <hip/hip_runtime.h>

typedef __attribute__((ext_vector_type(2))) float v2f;
typedef __attribute__((ext_vector_type(8))) float v8f;

#define PSTRIDE 72   // A-panel channel stride (banks: 8*c + row -> conflict-free)

// out[b,i,h,w] = sum_{c,j,k2} A(b,h,w,c,j,k2) * W[c,j,k2,i]
// GEMM: M = 1024*49 (per-image 49 rows padded to 64), N = 512, K = 9*128
// K order: kk = (j*3+k2)*128 + c  -> W row = c*9 + jk
__global__ __launch_bounds__(256, 1) void shiftconv_wmma(
    const float* __restrict__ x,    // (1024,256,7,7)
    const float* __restrict__ Wt,   // (128,3,3,512)
    float* __restrict__ out)        // (1024,512,7,7)
{
  __shared__ float xs[256 * 49];        // 49 KB: full x[b]
  __shared__ float ap[128 * PSTRIDE];   // 36 KB: A panel for current jk

  const int b   = blockIdx.x;
  const int tid = threadIdx.x;

  const float* xb = x + (size_t)b * (256 * 49);
  #pragma unroll 1
  for (int i = tid; i < 256 * 49; i += 256) xs[i] = xb[i];

  const int lane = tid & 31;
  const int wid  = tid >> 5;   // 8 waves
  const int half = lane >> 4;  // K-half for A/B fragments
  const int r16  = lane & 15;  // M row (A) / N col (B)
  const int nb   = wid * 64;   // this wave's N range

  // Build-phase geometry: this thread owns panel row `prow` (rows >48 clamp to 48)
  const int prow = tid & 63;
  const int cb   = tid >> 6;   // channel start (c = cb + 4k)
  int pm = prow > 48 ? 48 : prow;
  const int ph  = pm / 7;
  const int pw  = pm - ph * 7;
  const int ph0 = (ph + 6) % 7;   // (h-1) mod 7

  v8f acc[4][4];
  #pragma unroll
  for (int t = 0; t < 4; ++t)
    #pragma unroll
    for (int g = 0; g < 4; ++g)
      #pragma unroll
      for (int v = 0; v < 8; ++v)
        acc[t][g][v] = 0.0f;

  __syncthreads();   // xs ready

  #pragma unroll 1
  for (int jk = 0; jk < 9; ++jk) {
    const int j  = jk / 3;
    const int k2 = jk - j * 3;

    // ---- build A panel for this jk (each thread: 1 row x 32 channels) ----
    {
      int  wp = pw + k2;                         // w + k2
      bool iw = (wp >= 1) & (wp <= 7);           // W-dim pad mask
      int  ws = (wp + 5) % 7;                    // (w + k2 - 2) mod 7
      int  ha = ph + j;
      float f1 = (iw & (ha >= 1) & (ha <= 7)) ? 1.0f : 0.0f;
      int a1 = ha - 1; a1 = a1 < 0 ? 0 : (a1 > 6 ? 6 : a1);
      int bA = a1 * 7 + ws;
      int  hb = ph0 + j;
      float f2 = (iw & (hb >= 1) & (hb <= 7)) ? 1.0f : 0.0f;
      int a2 = hb - 1; a2 = a2 < 0 ? 0 : (a2 > 6 ? 6 : a2);
      int bB = 128 * 49 + a2 * 7 + ws;

      #pragma unroll 4
      for (int k = 0; k < 32; ++k) {
        int c = cb + 4 * k;
        ap[c * PSTRIDE + prow] = xs[c * 49 + bA] * f1 + xs[c * 49 + bB] * f2;
      }
    }
    __syncthreads();   // panel ready

    // ---- main GEMM sweep over this jk's 128 channels ----
    const int wrj = jk * 512 + nb + r16;   // W flat col base for this jk
    #pragma unroll 2
    for (int c0 = 0; c0 < 128; c0 += 4) {  // K step of 4
      const int cA = c0 + half * 2;

      // A fragments: mask-free panel reads (one ds_load_2addr per tile)
      v2f afr[4];
      const int pa = cA * PSTRIDE + r16;
      #pragma unroll
      for (int t = 0; t < 4; ++t) {
        afr[t].x = ap[pa + t * 16];
        afr[t].y = ap[pa + PSTRIDE + t * 16];
      }

      // B fragments: W row = cA*9 + jk
      v2f bfr[4];
      const int wr = cA * 9 * 512 + wrj;
      #pragma unroll
      for (int g = 0; g < 4; ++g) {
        bfr[g].x = Wt[wr + g * 16];
        bfr[g].y = Wt[wr + 4608 + g * 16];   // +9*512 -> channel cA+1
      }

      #pragma unroll
      for (int t = 0; t < 4; ++t)
        #pragma unroll
        for (int g = 0; g < 4; ++g)
          acc[t][g] = __builtin_amdgcn_wmma_f32_16x16x4_f32(
              false, afr[t], false, bfr[g], (short)0, acc[t][g],
              false, false);
    }
    __syncthreads();   // done reading panel before next jk overwrites it
  }

  // Epilogue: D vgpr v holds (M = t*16 + half*8 + v, N = nb + g*16 + r16)
  #pragma unroll
  for (int t = 0; t < 4; ++t) {
    #pragma unroll
    for (int g = 0; g < 4; ++g) {
      const int col = nb + g * 16 + r16;
      float* op = out + ((size_t)b * 512 + col) * 49;
      #pragma unroll
      for (int v = 0; v < 8; ++v) {
        int mrow = t * 16 + half * 8 + v;
        if (mrow < 49) op[mrow] = acc[t][g][v];
      }
    }
  }
}

extern "C" void kernel_launch(void* const* d_in, const int* in_sizes, int n_in,
                              void* d_out, int out_size, void* d_ws, size_t ws_size,
                              hipStream_t stream) {
  (void)in_sizes; (void)n_in; (void)out_size; (void)d_ws; (void)ws_size;
  const float* x = (const float*)d_in[0];   // 1024*256*7*7 fp32
  const float* W = (const float*)d_in[1];   // 128*3*3*512 fp32
  float* out = (float*)d_out;               // 1024*512*7*7 fp32
  shiftconv_wmma<<<1024, 256, 0, stream>>>(x, W, out);
}